// Att_score_42984032698908
// MI455X (gfx1250) — compile-verified
//
#include <hip/hip_runtime.h>
#include <hip/hip_bf16.h>
#include <math.h>

typedef __attribute__((ext_vector_type(2))) float v2f;
typedef __attribute__((ext_vector_type(8))) float v8f;

#define NEG_SLOPE 0.01f

// ---------------------------------------------------------------------------
// Phase 1: q/k GEMV pair as WMMA GEMM  (16384 x 256) x (256 x 2)
// One wave per 16-row tile. A = 16x4 f32 tile of i_em, B = 4x16 f32 with
// column 0 = aq chunk, column 1 = ak chunk, rest zero.
// A layout (ISA 7.12.2, 32-bit A 16x4): lanes 0-15 = M, lanes 16-31 = M with
// K offset +2; VGPR0 = K even slot, VGPR1 = K odd slot.
// B-fragment loads are branch-free: lanes with n>=2 load a clamped (valid)
// address and zero the value with a 0/1 mask multiply, so EXEC stays all-1s
// through the whole WMMA loop (no s_and_saveexec serialization).
// ---------------------------------------------------------------------------
__global__ __launch_bounds__(256) void qk_wmma_kernel(
    const float* __restrict__ i_em,   // (BN, 256) row-major
    const float* __restrict__ a_w,    // (512,)  [0:256)=aq, [256:512)=ak
    float* __restrict__ qk,           // q at [0, BN), k at [BN, 2*BN)
    int BN)
{
  const int lane  = threadIdx.x & 31;
  const int wave  = threadIdx.x >> 5;
  const int tile  = blockIdx.x * 8 + wave;       // 16-row tile
  const int rowBase = tile * 16;
  if (rowBase >= BN) return;

  const int m     = lane & 15;                   // row (A) / column (B,C)
  const int khalf = lane >> 4;                   // 0 or 1

  const float* arow  = i_em + (size_t)(rowBase + m) * 256 + (size_t)(khalf * 2);
  // Clamp weight row for lanes n>=2 to a valid address; zero via mask.
  const int   wrow   = (m < 2) ? m : 0;
  const float bmask  = (m < 2) ? 1.0f : 0.0f;
  const float* wbase = a_w + wrow * 256 + khalf * 2;

  v8f acc = {};

#pragma unroll 4
  for (int kbase = 0; kbase < 256; kbase += 4) {
    // A fragment: this lane holds (K = khalf*2, khalf*2+1) of row m
    float2 av = *(const float2*)(arow + kbase);
    v2f a; a.x = av.x; a.y = av.y;

    // B fragment: lane holds B[K = khalf*2 + vgpr][N = m], branch-free
    float2 wv = *(const float2*)(wbase + kbase);
    v2f bfrag; bfrag.x = wv.x * bmask; bfrag.y = wv.y * bmask;

    acc = __builtin_amdgcn_wmma_f32_16x16x4_f32(
        /*neg_a=*/false, a, /*neg_b=*/false, bfrag,
        /*c_mod=*/(short)0, acc, /*reuse_a=*/false, /*reuse_b=*/false);
  }

  // D[mrow][n]: vgpr = mrow%8, lane = n + 16*(mrow/8).
  // q = column n=0  -> lanes 0 (rows 0-7) and 16 (rows 8-15)
  // k = column n=1  -> lanes 1 (rows 0-7) and 17 (rows 8-15)
  float* qout = qk;
  float* kout = qk + BN;
  if (lane == 0) {
#pragma unroll
    for (int i = 0; i < 8; ++i) qout[rowBase + i] = acc[i];
  } else if (lane == 16) {
#pragma unroll
    for (int i = 0; i < 8; ++i) qout[rowBase + 8 + i] = acc[i];
  } else if (lane == 1) {
#pragma unroll
    for (int i = 0; i < 8; ++i) kout[rowBase + i] = acc[i];
  } else if (lane == 17) {
#pragma unroll
    for (int i = 0; i < 8; ++i) kout[rowBase + 8 + i] = acc[i];
  }
}

// ---------------------------------------------------------------------------
// Phase 2: per-row leaky-relu + softmax over j (1024 wide), one block per row.
// 256 threads x 4 elements; wave32 shuffle + LDS reductions.
// This phase is the HBM-store-bound part (64 MB of output).
// ---------------------------------------------------------------------------
__global__ __launch_bounds__(256) void leaky_softmax_kernel(
    const float* __restrict__ qk,     // q[0,BN), k[BN,2BN)
    const float* __restrict__ a_b,    // (1,)
    float* __restrict__ out,          // (BN, 1024)
    int BN)
{
  const int row  = blockIdx.x;        // b*1024 + i
  const int b    = row >> 10;
  const int tid  = threadIdx.x;
  const int lane = tid & 31;
  const int wave = tid >> 5;

  const float bias = a_b[0];
  const float qv   = qk[row];
  const float* krow = qk + BN + (size_t)b * 1024;

  float s[4];
  float mx = -INFINITY;
#pragma unroll
  for (int u = 0; u < 4; ++u) {
    const int j = tid + u * 256;
    float v = qv + krow[j] + bias;
    v = (v >= 0.0f) ? v : NEG_SLOPE * v;
    s[u] = v;
    mx = fmaxf(mx, v);
  }

  // --- max reduction: wave32 shuffle-xor then cross-wave via LDS ---
  __shared__ float redmax[8];
  __shared__ float redsum[8];
#pragma unroll
  for (int off = 16; off > 0; off >>= 1)
    mx = fmaxf(mx, __shfl_xor(mx, off, 32));
  if (lane == 0) redmax[wave] = mx;
  __syncthreads();
  mx = redmax[0];
#pragma unroll
  for (int w = 1; w < 8; ++w) mx = fmaxf(mx, redmax[w]);

  // --- exp + sum reduction ---
  float sum = 0.0f;
#pragma unroll
  for (int u = 0; u < 4; ++u) {
    s[u] = __expf(s[u] - mx);
    sum += s[u];
  }
#pragma unroll
  for (int off = 16; off > 0; off >>= 1)
    sum += __shfl_xor(sum, off, 32);
  if (lane == 0) redsum[wave] = sum;
  __syncthreads();
  sum = 0.0f;
#pragma unroll
  for (int w = 0; w < 8; ++w) sum += redsum[w];

  const float inv = 1.0f / sum;
  float* orow = out + (size_t)row * 1024;
#pragma unroll
  for (int u = 0; u < 4; ++u)
    orow[tid + u * 256] = s[u] * inv;   // coalesced 64 MB stream
}

// ---------------------------------------------------------------------------
extern "C" void kernel_launch(void* const* d_in, const int* in_sizes, int n_in,
                              void* d_out, int out_size, void* d_ws, size_t ws_size,
                              hipStream_t stream) {
  const float* i_em = (const float*)d_in[0];   // (16,1024,256) f32
  const float* a_w  = (const float*)d_in[1];   // (512,) f32
  const float* a_b  = (const float*)d_in[2];   // (1,)  f32
  float* out = (float*)d_out;                  // (16,1024,1024,1) f32
  float* qk  = (float*)d_ws;                   // 2*BN floats = 128 KB scratch

  const int D  = 256;
  const int BN = in_sizes[0] / D;              // 16384

  // Phase 1: 1024 tiles of 16 rows, 8 waves (one tile each) per block.
  const int tiles  = BN / 16;
  const int blocks = (tiles + 7) / 8;
  qk_wmma_kernel<<<blocks, 256, 0, stream>>>(i_em, a_w, qk, BN);

  // Phase 2: one block per output row.
  leaky_softmax_kernel<<<BN, 256, 0, stream>>>(qk, a_b, out, BN);
}